// conv_caps_EM_23596550324379
// MI455X (gfx1250) — compile-verified
//
#include <hip/hip_runtime.h>
#include <hip/hip_bf16.h>

// Problem constants (from reference)
#define NPOS  400    // b*oh*ow = 4*10*10
#define NI    288    // kk*ICAPS = 9*32
#define NOC   32     // OCAPS
#define OH    512    // OCAPS*OPOSE
#define CIN   544    // ICAPS*(IPOSE+1)
#define REF_EPS 1e-12f
#define REF_LAMBDA 1e-3f

typedef float v2f __attribute__((ext_vector_type(2)));
typedef float v8f __attribute__((ext_vector_type(8)));

// ---------------------------------------------------------------------------
// Kernel 1: votes via V_WMMA_F32_16X16X4_F32  (exact 4x4-matmul engine)
//
// v[n,i,o,r,c] = sum_k p[n,i,r,k] * W[i,o,k,c]
// WMMA mapping (per wave, fixed input capsule i, fixed 16-position tile):
//   A[16 x 4]  : M = 16 positions n,  K = pose inner index k (pose row r fixed)
//   B[4 x 16]  : K = k,               N = (o_sub within o-block, c)
//   D[16 x 16] : votes for 16 positions x (4 ocaps x 4 cols), f32 accum
// All 4 A fragments and all 8 B fragments are preloaded (one load clause, one
// wait), then the 8 o-blocks x 4 rows = 32 WMMAs run fully unrolled with no
// vmem waits in the matrix pipeline. D is staged through LDS per o-block so
// the f16 global writes of v[n][i][o][h] are coalesced 16B-per-lane stores
// (v stored f16: 118 MB total -> resident in the 192 MB L2 for the EM passes).
// Grid: (25 tiles, 36 i-groups), 256 threads = 8 waves, wave w -> i = ig*8+w.
// ---------------------------------------------------------------------------
__global__ __launch_bounds__(256) void caps_votes_wmma(
    const float* __restrict__ x, const float* __restrict__ Wa,
    _Float16* __restrict__ v)
{
  __shared__ _Float16 stage[8][16][64];  // per-wave staging for one o-block

  const int wave = threadIdx.x >> 5;
  const int lane = threadIdx.x & 31;
  const int tile = blockIdx.x;                 // 0..24  (16 positions each)
  const int i    = blockIdx.y * 8 + wave;      // 0..287 input capsule

  const int ky = i / 96, kx = (i / 32) % 3, icap = i & 31;

  // ---- A fragments (32-bit A 16x4 layout, ISA 7.12.2):
  // lane L: row M = L%16; lanes 0-15 hold K=0 (v0), K=1 (v1); lanes 16-31 K=2,3
  const int m   = lane & 15;
  const int n   = tile * 16 + m;               // global position, < 400
  const int bb  = n / 100, rem = n % 100;
  const int py  = rem / 10, pxc = rem % 10;
  const int yy  = py + ky - 1, xx = pxc + kx - 1;
  const bool inb = (yy >= 0) && (yy < 10) && (xx >= 0) && (xx < 10);
  const int k0  = (lane < 16) ? 0 : 2;
  const float* px = x + (size_t)((bb * 10 + yy) * 10 + xx) * CIN + icap * 17;

  v2f A[4];
#pragma unroll
  for (int r = 0; r < 4; ++r) {
    A[r].x = inb ? px[r * 4 + k0]     : 0.0f;
    A[r].y = inb ? px[r * 4 + k0 + 1] : 0.0f;
  }

  // ---- B fragments: lane column N = (o_sub, c); preload ALL 8 o-blocks so
  // the loads clause together and the WMMA stream never waits on vmem.
  const int col  = lane & 15;
  const int osub = col >> 2, cc = col & 3;

  v2f B[8];
#pragma unroll
  for (int ob = 0; ob < 8; ++ob) {
    const int o = ob * 4 + osub;
    const float* pw = Wa + ((size_t)(i * NOC + o) * 4 + k0) * 4 + cc;
    B[ob].x = pw[0];   // k = k0
    B[ob].y = pw[4];   // k = k0+1 (stride 4 floats between k rows)
  }

  const int mbase = (lane < 16) ? 0 : 8;
  const int nl    = lane >> 1, half = lane & 1;   // for the coalesced write
  const int gn    = tile * 16 + nl;

#pragma unroll
  for (int ob = 0; ob < 8; ++ob) {
#pragma unroll
    for (int r = 0; r < 4; ++r) {
      v8f C = {};
      v8f D = __builtin_amdgcn_wmma_f32_16x16x4_f32(
          /*neg_a=*/false, A[r], /*neg_b=*/false, B[ob],
          /*c_mod=*/(short)0, C, /*reuse_a=*/false, /*reuse_b=*/false);
      // D layout: VGPR j -> M=j (lanes 0-15) / M=j+8 (lanes 16-31); N = lane%16
#pragma unroll
      for (int j = 0; j < 8; ++j)
        stage[wave][mbase + j][osub * 16 + r * 4 + cc] = (_Float16)D[j];
    }
    __syncthreads();

    // Coalesced write of this o-block: 16 n x 64 f16 (128 B per n) per wave.
    // 2 lanes per position, 4 x 16-byte transfers each.
    {
      _Float16*       dst = v + ((size_t)gn * NI + i) * OH + ob * 64 + half * 32;
      const _Float16* src = &stage[wave][nl][half * 32];
#pragma unroll
      for (int j = 0; j < 4; ++j)
        ((float4*)dst)[j] = ((const float4*)src)[j];
    }
    __syncthreads();
  }
}

// ---------------------------------------------------------------------------
// Kernel 2: EM routing, one workgroup per spatial position (400 WGs).
// v chunk per position = 288 KB f16, L2-resident; reads are coalesced.
// r[288][32] + stats live in LDS (~45 KB).
// ---------------------------------------------------------------------------
__global__ __launch_bounds__(256) void caps_em_routing(
    const float* __restrict__ x, const _Float16* __restrict__ v,
    const float* __restrict__ beta_u, const float* __restrict__ beta_a,
    float* __restrict__ out)
{
  __shared__ float r_sh[NI * NOC];                 // 36 KB
  __shared__ float a_in[NI];
  __shared__ float mu[OH], inv2sig[OH], logsig[OH];
  __shared__ float r_sum[NOC], inv_rsum[NOC], a_out_sh[NOC], ln_a[NOC];

  const int n = blockIdx.x;
  const int t = threadIdx.x;
  const _Float16* vn = v + (size_t)n * NI * OH;
  const int bb = n / 100, rem = n % 100;
  const int py = rem / 10, pxc = rem % 10;
  const float ln2pi = 1.8378770664093453f;

  // load input activations a_in[i]
  for (int i = t; i < NI; i += 256) {
    const int ky = i / 96, kx = (i / 32) % 3, icap = i & 31;
    const int yy = py + ky - 1, xx = pxc + kx - 1;
    const bool inb = (yy >= 0) && (yy < 10) && (xx >= 0) && (xx < 10);
    a_in[i] = inb ? x[(size_t)((bb * 10 + yy) * 10 + xx) * CIN + icap * 17 + 16]
                  : 0.0f;
  }
  for (int p = t; p < NI * NOC; p += 256) r_sh[p] = 1.0f / 32.0f;
  __syncthreads();

  for (int it = 0; it < 3; ++it) {
    // ---- A: rr = r * a_in, normalized over o (in place)
    for (int i = t; i < NI; i += 256) {
      const float ai = a_in[i];
      float s = 0.0f;
      for (int o = 0; o < NOC; ++o) s += r_sh[i * NOC + o];
      const float scale = ai / (s * ai + REF_EPS);
      for (int o = 0; o < NOC; ++o) r_sh[i * NOC + o] *= scale;
    }
    if (t < NOC) r_sum[t] = 0.0f;
    __syncthreads();

    // ---- B: r_sum[o] = sum_i rr[i][o]   (LDS float atomics = ds_add_f32)
    {
      const int o = t & 31;
      float acc = 0.0f;
      for (int i = t >> 5; i < NI; i += 8) acc += r_sh[i * NOC + o];
      atomicAdd(&r_sum[o], acc);
    }
    __syncthreads();
    if (t < NOC) inv_rsum[t] = 1.0f / (r_sum[t] + REF_EPS);
    __syncthreads();

    // ---- C1: mu[o][h] = sum_i coeff[i][o] * v[i][o][h]   (coalesced f16 reads)
    for (int p = t; p < OH; p += 256) {
      const int o = p >> 4;
      const float inv = inv_rsum[o];
      float acc = 0.0f;
      for (int i = 0; i < NI; ++i)
        acc += r_sh[i * NOC + o] * inv * (float)vn[i * OH + p];
      mu[p] = acc;
    }
    __syncthreads();

    // ---- C2: sigma_sq, and derived 1/(2s2), log(sqrt(s2))
    for (int p = t; p < OH; p += 256) {
      const int o = p >> 4;
      const float inv = inv_rsum[o];
      const float mm = mu[p];
      float acc = 0.0f;
      for (int i = 0; i < NI; ++i) {
        const float d = (float)vn[i * OH + p] - mm;
        acc += r_sh[i * NOC + o] * inv * d * d;
      }
      const float s2 = acc + REF_EPS;
      inv2sig[p] = 0.5f / s2;
      logsig[p]  = 0.5f * __logf(s2);
    }
    __syncthreads();

    // ---- D: a_out[o] = sigmoid(lambda*(beta_a - sum_h cost_h))
    if (t < NOC) {
      float ch = 0.0f;
      for (int h = 0; h < 16; ++h)
        ch += beta_u[t] + logsig[t * 16 + h] + REF_EPS;
      ch *= r_sum[t];
      const float ao = 1.0f / (1.0f + __expf(-REF_LAMBDA * (beta_a[t] - ch)));
      a_out_sh[t] = ao;
      ln_a[t] = __logf(ao);
    }
    __syncthreads();

    // ---- E: r = softmax_o( sum_h ln_p + ln a_out )
    if (it < 2) {
      for (int i = t; i < NI; i += 256) {
        float lnap[NOC];
        float mx = -3.4e38f;
#pragma unroll 4
        for (int o = 0; o < NOC; ++o) {
          float s = 0.0f;
          for (int h = 0; h < 16; ++h) {
            const int p = o * 16 + h;
            const float d = (float)vn[i * OH + p] - mu[p];
            s += -d * d * inv2sig[p] - logsig[p] - 0.5f * ln2pi;
          }
          s += ln_a[o];
          lnap[o] = s;
          mx = fmaxf(mx, s);
        }
        float se = 0.0f;
        for (int o = 0; o < NOC; ++o) { lnap[o] = __expf(lnap[o] - mx); se += lnap[o]; }
        const float invs = 1.0f / se;
        for (int o = 0; o < NOC; ++o) r_sh[i * NOC + o] = lnap[o] * invs;
      }
      __syncthreads();
    }
  }

  // ---- output: [mu (512) | a_out (32)] per position
  for (int p = t; p < OH; p += 256) out[(size_t)n * CIN + p] = mu[p];
  if (t < NOC) out[(size_t)n * CIN + OH + t] = a_out_sh[t];
}

// ---------------------------------------------------------------------------
extern "C" void kernel_launch(void* const* d_in, const int* in_sizes, int n_in,
                              void* d_out, int out_size, void* d_ws, size_t ws_size,
                              hipStream_t stream) {
  const float* x      = (const float*)d_in[0];   // 4*10*10*544
  const float* Wa     = (const float*)d_in[1];   // 288*32*4*4
  const float* beta_u = (const float*)d_in[2];   // 32
  const float* beta_a = (const float*)d_in[3];   // 32
  float* out = (float*)d_out;                    // 4*10*10*544
  _Float16* v = (_Float16*)d_ws;                 // 400*288*512 f16 = ~118 MB (fits L2)

  caps_votes_wmma<<<dim3(25, 36), 256, 0, stream>>>(x, Wa, v);
  caps_em_routing<<<400, 256, 0, stream>>>(x, v, beta_u, beta_a, out);
}